// LengthRegulator_21268678049844
// MI455X (gfx1250) — compile-verified
//
#include <hip/hip_runtime.h>
#include <hip/hip_bf16.h>

typedef float v2f __attribute__((ext_vector_type(2)));
typedef float v4f __attribute__((ext_vector_type(4)));
typedef float v8f __attribute__((ext_vector_type(8)));

#define BB 4
#define SS 512
#define DD 512
#define NN 256   // D/2

// ---------------------------------------------------------------------------
// 1) Inclusive scan of durations[0] (512 ints), one block.
// ---------------------------------------------------------------------------
__global__ void scan_kernel(const int* __restrict__ dur, int* __restrict__ csum) {
    __shared__ int sh[SS];
    int i = threadIdx.x;
    sh[i] = dur[i];
    __syncthreads();
    for (int off = 1; off < SS; off <<= 1) {
        int v = (i >= off) ? sh[i - off] : 0;
        __syncthreads();
        sh[i] += v;
        __syncthreads();
    }
    csum[i] = sh[i];
}

// ---------------------------------------------------------------------------
// 2) Per-frame mapping: searchsorted(csum, t, 'right'), alpha, validity (ic=-1).
// ---------------------------------------------------------------------------
__global__ void map_kernel(const int* __restrict__ csum, const int* __restrict__ dur0,
                           int* __restrict__ ic_out, float* __restrict__ alpha_out, int T) {
    __shared__ int sc[SS];
    for (int i = threadIdx.x; i < SS; i += blockDim.x) sc[i] = csum[i];
    __syncthreads();
    int t = blockIdx.x * blockDim.x + threadIdx.x;
    if (t >= T) return;
    int L = sc[SS - 1];
    if (t >= L) { ic_out[t] = -1; alpha_out[t] = 0.0f; return; }
    int lo = 0, hi = SS;                       // first idx with csum[idx] > t
    while (lo < hi) { int mid = (lo + hi) >> 1; if (sc[mid] <= t) lo = mid + 1; else hi = mid; }
    int ic = lo < (SS - 1) ? lo : (SS - 1);
    int d  = dur0[ic];
    int j  = t - (sc[ic] - d);
    float alpha = (ic == SS - 1) ? 0.0f : (float)j / (float)d;
    ic_out[t] = ic;
    alpha_out[t] = alpha;
}

// ---------------------------------------------------------------------------
// 3) Expansion: pure-bandwidth 128-bit interpolation. One block per (b, t) row.
//    394 MB of write-once stores (> 192 MB L2) -> non-temporal store hint so
//    the stream rinses to HBM instead of evicting the hot x/W1 working set.
//    x rows (4 MB total, each reused ~23x) stay regular-temporal / L2-resident.
//    ~17 us floor at 23.3 TB/s HBM.
// ---------------------------------------------------------------------------
__global__ void expand_kernel(const float* __restrict__ x, const int* __restrict__ ic_arr,
                              const float* __restrict__ alpha_arr, float* __restrict__ out, int T) {
    int bt = blockIdx.x;
    int b = bt / T, t = bt - b * T;
    v4f* o = (v4f*)(out + (size_t)bt * DD);
    int d4 = threadIdx.x;                      // 0..127, DD/4 = 128
    int ic = ic_arr[t];
    if (ic < 0) {
        v4f z = {0.f, 0.f, 0.f, 0.f};
        __builtin_nontemporal_store(z, o + d4);
        return;
    }
    float a = alpha_arr[t];
    int ir = (ic + 1 < SS) ? ic + 1 : SS - 1;
    const v4f* l = (const v4f*)(x + ((size_t)b * SS + ic) * DD);
    const v4f* r = (const v4f*)(x + ((size_t)b * SS + ir) * DD);
    v4f lv = l[d4], rv = r[d4];
    v4f res = lv + a * (rv - lv);              // (1-a)*left + a*right
    __builtin_nontemporal_store(res, o + d4);
}

// ---------------------------------------------------------------------------
// 4) h = relu(x @ W1 + b1) via V_WMMA_F32_16X16X4_F32.
//    M=B*S=2048, K=512, N=256. One wave per 16x16 C tile; 128 k-steps of 16x16x4.
//    A 16x4 f32: lane l -> row m=l%16, regs K = 2*(l/16)+{0,1}  (contiguous float2)
//    B 4x16  f32: lane l -> col n=l%16, reg r -> K = 2*(l/16)+r
//    C/D 16x16: reg j, lane l -> (m = j + 8*(l/16), n = l%16)
//    EXEC all-ones: 256 blocks x 8 waves == 2048 tiles exactly.
// ---------------------------------------------------------------------------
__global__ __launch_bounds__(256) void mlp1_wmma_kernel(const float* __restrict__ x,
                                                        const float* __restrict__ W1,
                                                        const float* __restrict__ b1,
                                                        float* __restrict__ h) {
    const int K = DD;
    int wave  = blockIdx.x * (blockDim.x >> 5) + (threadIdx.x >> 5);
    int mtile = wave >> 4;          // 0..127
    int ntile = wave & 15;          // 0..15
    int lane  = threadIdx.x & 31;
    int half  = lane >> 4;
    int lm    = lane & 15;

    const float* arow = x  + (size_t)(mtile * 16 + lm) * K;   // A row for this lane
    const float* bcol = W1 + ntile * 16 + lm;                 // B column for this lane

    v8f acc = {};
    #pragma unroll 4
    for (int k = 0; k < K; k += 4) {
        int ka = k + half * 2;
        v2f a; a.x = arow[ka];            a.y = arow[ka + 1];
        v2f b; b.x = bcol[(size_t)ka * NN]; b.y = bcol[(size_t)(ka + 1) * NN];
        acc = __builtin_amdgcn_wmma_f32_16x16x4_f32(
            /*neg_a=*/false, a, /*neg_b=*/false, b,
            /*c_mod=*/(short)0, acc, /*reuse_a=*/false, /*reuse_b=*/false);
    }

    float bias = b1[ntile * 16 + lm];     // bias depends on column n only
    #pragma unroll
    for (int j = 0; j < 8; ++j) {
        float v = acc[j] + bias;
        v = v > 0.0f ? v : 0.0f;          // relu
        int m = j + 8 * half;
        h[(size_t)(mtile * 16 + m) * NN + ntile * 16 + lm] = v;
    }
}

// ---------------------------------------------------------------------------
// 5) dur_mlp = softplus(h @ W2 + b2): one wave per row (2048 rows), shuffle-reduce.
// ---------------------------------------------------------------------------
__global__ __launch_bounds__(256) void mlp2_kernel(const float* __restrict__ h,
                                                   const float* __restrict__ W2,
                                                   const float* __restrict__ b2,
                                                   float* __restrict__ dur_out) {
    int row  = blockIdx.x * (blockDim.x >> 5) + (threadIdx.x >> 5);
    int lane = threadIdx.x & 31;
    const float* hr = h + (size_t)row * NN;
    float s = 0.0f;
    #pragma unroll
    for (int j = 0; j < NN / 32; ++j) {
        int idx = lane + j * 32;
        s += hr[idx] * W2[idx];
    }
    #pragma unroll
    for (int off = 16; off > 0; off >>= 1) s += __shfl_down(s, off, 32);
    if (lane == 0) {
        float v = s + b2[0];
        // numerically stable softplus: max(v,0) + log1p(exp(-|v|))
        dur_out[row] = fmaxf(v, 0.0f) + log1pf(expf(-fabsf(v)));
    }
}

// ---------------------------------------------------------------------------
extern "C" void kernel_launch(void* const* d_in, const int* in_sizes, int n_in,
                              void* d_out, int out_size, void* d_ws, size_t ws_size,
                              hipStream_t stream) {
    const float* x         = (const float*)d_in[0];
    const float* W1        = (const float*)d_in[1];
    const float* b1        = (const float*)d_in[2];
    const float* W2        = (const float*)d_in[3];
    const float* b2        = (const float*)d_in[4];
    const int*   durations = (const int*)d_in[5];   // (B,S); row 0 is dur0

    // total_length derived host-side from out_size: out = B*T*D (expanded) + B*S (dur_mlp)
    const int T = (out_size - BB * SS) / (BB * DD);

    // workspace layout
    char* ws = (char*)d_ws;
    size_t off = 0;
    int*   csum  = (int*)(ws + off);   off += 4096;                         // 512 ints, padded
    int*   ic    = (int*)(ws + off);   off += ((size_t)T * 4 + 255) & ~255ull;
    float* alpha = (float*)(ws + off); off += ((size_t)T * 4 + 255) & ~255ull;
    float* h     = (float*)(ws + off);                                      // 2048*256 f32

    float* out_exp = (float*)d_out;
    float* out_dur = out_exp + (size_t)BB * T * DD;

    scan_kernel<<<1, SS, 0, stream>>>(durations, csum);
    map_kernel<<<(T + 255) / 256, 256, 0, stream>>>(csum, durations, ic, alpha, T);
    expand_kernel<<<BB * T, DD / 4, 0, stream>>>(x, ic, alpha, out_exp, T);
    mlp1_wmma_kernel<<<(BB * SS / 16) * (NN / 16) / 8, 256, 0, stream>>>(x, W1, b1, h);
    mlp2_kernel<<<BB * SS / 8, 256, 0, stream>>>(h, W2, b2, out_dur);
}